// BandSplitModule_49331994362210
// MI455X (gfx1250) — compile-verified
//
#include <hip/hip_runtime.h>
#include <hip/hip_bf16.h>

typedef __attribute__((ext_vector_type(2))) float v2f;
typedef __attribute__((ext_vector_type(8))) float v8f;

#define NBANDS 43
#define FC 128
#define B_ 8
#define C_ 2
#define F_ 1025
#define T_ 517
#define EPS_ 1e-5f
#define MROWS (B_ * T_)          // 4136 rows of the (b,t) GEMM
#define MTILE 4                  // 4 x (16-row) M tiles per block -> 64 rows

__constant__ int g_band_s[NBANDS] = {
    0, 10, 20, 30, 40, 50, 60, 70, 80, 90,
    100, 110, 120, 130, 140, 150, 160, 170, 180, 190,
    200, 220, 240, 260, 280, 300, 320, 340, 360, 380,
    400, 440, 480, 520, 560, 600, 640, 680,
    720, 781, 842, 903, 964};

struct BandArgs {
    const float* W[NBANDS];
    const float* bias[NBANDS];
};

// ---------------- Pass 1a: per-(band, batch) stats -> inv, mu*inv ----------
// W = band width (compile-time). Grid: (B_, #bands in group); band = B0+blockIdx.y
template <int W>
__global__ __launch_bounds__(256) void band_stats(const float* __restrict__ x,
                                                  float* __restrict__ inv,
                                                  float* __restrict__ invmu,
                                                  int bandBase) {
    const int b = blockIdx.x;
    const int band = bandBase + blockIdx.y;
    const int s = g_band_s[band];
    constexpr int chunkLen = T_ * 2;   // contiguous (t, reim) run per (c, freq)
    constexpr int chunks = 2 * W;      // c * freq
    constexpr int total = chunks * chunkLen;

    float sum = 0.f, sq = 0.f;
    for (int i = threadIdx.x; i < total; i += 256) {
        int chunk = i / chunkLen;
        int off = i - chunk * chunkLen;
        int c = chunk / W;
        int fr = chunk - c * W;
        float v = x[(((size_t)b * C_ + c) * F_ + (s + fr)) * (size_t)chunkLen + off];
        sum += v;
        sq += v * v;
    }
    __shared__ float ssum[256];
    __shared__ float ssq[256];
    ssum[threadIdx.x] = sum;
    ssq[threadIdx.x] = sq;
    __syncthreads();
    for (int st = 128; st > 0; st >>= 1) {
        if (threadIdx.x < st) {
            ssum[threadIdx.x] += ssum[threadIdx.x + st];
            ssq[threadIdx.x] += ssq[threadIdx.x + st];
        }
        __syncthreads();
    }
    if (threadIdx.x == 0) {
        float N = (float)total;
        float mu = ssum[0] / N;
        float var = ssq[0] / N - mu * mu;
        float iv = rsqrtf(var + EPS_);
        inv[band * B_ + b] = iv;
        invmu[band * B_ + b] = mu * iv;
    }
}

// ---------------- Pass 1b: per-band W row sums (for mean folding) ----------
template <int W>
__global__ __launch_bounds__(128) void band_rowsum(BandArgs args,
                                                   float* __restrict__ rowsum,
                                                   int bandBase) {
    const int band = bandBase + blockIdx.x;
    const int o = threadIdx.x;
    constexpr int K = W * 4;
    const float* __restrict__ Wb = args.W[band];
    float acc = 0.f;
#pragma unroll 4
    for (int f = 0; f < K; ++f) acc += Wb[(size_t)o * K + f];
    rowsum[band * FC + o] = acc;
}

// ---------------- Pass 2: WMMA GEMM with fused layernorm epilogue ----------
// grid.x = 64-row M blocks, grid.y = #bands in this width group.
// 256 threads = 8 waves; wave i owns output columns [16i, 16i+16) and all
// four 16-row M tiles (register blocking: 1 B-frag feeds 4 WMMAs).
template <int W>
__global__ __launch_bounds__(256) void band_gemm(const float* __restrict__ x,
                                                 BandArgs args,
                                                 const float* __restrict__ inv,
                                                 const float* __restrict__ invmu,
                                                 const float* __restrict__ rowsum,
                                                 float* __restrict__ out,
                                                 int bandBase) {
    constexpr int K = W * 4;                   // multiple of 4
    constexpr int Kpad = K + 2;                // 8B-aligned b64 LDS, bank-safe stride
    const int mb = blockIdx.x;                 // 64-row block index
    const int band = bandBase + blockIdx.y;
    const int s = g_band_s[band];
    const int row0 = mb * (16 * MTILE);

    __shared__ float a_lds[16 * MTILE * Kpad]; // <= 62,976 B for W=61

    const int tid = threadIdx.x;

    // Cooperative A-tile load (64 rows x K). k = ((c*2+reim)*W + fr) is
    // wave-uniform (m = tid&63 varies within a wave) and the decode is all
    // compile-time-constant arithmetic; x reads are stride-2-in-t coalesced.
    constexpr int total = 16 * MTILE * K;
    for (int i = tid; i < total; i += 256) {
        int m = i & 63;
        int k = i >> 6;
        int row = row0 + m;
        float v = 0.f;
        if (row < MROWS) {
            int b = row / T_;
            int t = row - b * T_;
            int c = k / (2 * W);
            int rem = k - c * 2 * W;
            int reim = rem / W;
            int fr = rem - reim * W;
            v = x[((((size_t)b * C_ + c) * F_ + (s + fr)) * T_ + t) * 2 + reim];
        }
        a_lds[m * Kpad + k] = v;
    }
    __syncthreads();

    const int wave = tid >> 5;
    const int lane = tid & 31;
    const int lh = lane & 15;
    const int hi = lane >> 4;          // 0: K={0,1}, 1: K={2,3} (ISA 16x4 f32 layout)
    const int col = wave * 16 + lh;    // output column this lane supplies (B-matrix N)

    const float* __restrict__ Wb = args.W[band];
    const float* __restrict__ wrow = Wb + col * K + 2 * hi;
    const float* __restrict__ a0p = a_lds + (0 * 16 + lh) * Kpad + 2 * hi;
    const float* __restrict__ a1p = a_lds + (1 * 16 + lh) * Kpad + 2 * hi;
    const float* __restrict__ a2p = a_lds + (2 * 16 + lh) * Kpad + 2 * hi;
    const float* __restrict__ a3p = a_lds + (3 * 16 + lh) * Kpad + 2 * hi;

    __builtin_prefetch(wrow, 0, 0);    // global_prefetch_b8: warm W row in L0/L2

    v8f acc0 = {}, acc1 = {}, acc2 = {}, acc3 = {};
#pragma unroll 4
    for (int k = 0; k < K; k += 4) {
        v2f bf = *(const v2f*)(wrow + k);   // B(k+2hi.., n=lh) = W[col, k..]
        v2f a0 = *(const v2f*)(a0p + k);
        v2f a1 = *(const v2f*)(a1p + k);
        v2f a2 = *(const v2f*)(a2p + k);
        v2f a3 = *(const v2f*)(a3p + k);
        acc0 = __builtin_amdgcn_wmma_f32_16x16x4_f32(false, a0, false, bf,
                                                     (short)0, acc0, false, false);
        acc1 = __builtin_amdgcn_wmma_f32_16x16x4_f32(false, a1, false, bf,
                                                     (short)0, acc1, false, false);
        acc2 = __builtin_amdgcn_wmma_f32_16x16x4_f32(false, a2, false, bf,
                                                     (short)0, acc2, false, false);
        acc3 = __builtin_amdgcn_wmma_f32_16x16x4_f32(false, a3, false, bf,
                                                     (short)0, acc3, false, false);
    }

    // Epilogue: out = inv_b * acc - (mu_b*inv_b) * rowsumW[col] + bias[col]
    const float rs = rowsum[band * FC + col];
    const float bv = args.bias[band][col];
    v8f accs[MTILE] = {acc0, acc1, acc2, acc3};
#pragma unroll
    for (int mt = 0; mt < MTILE; ++mt) {
#pragma unroll
        for (int r = 0; r < 8; ++r) {
            int row = row0 + mt * 16 + r + 8 * hi;  // C/D layout: vgpr r -> M=r+8*hi
            if (row < MROWS) {
                int b = row / T_;
                int t = row - b * T_;
                float iv = inv[band * B_ + b];
                float ivmu = invmu[band * B_ + b];
                out[(((size_t)b * NBANDS + band) * T_ + t) * FC + col] =
                    iv * accs[mt][r] - ivmu * rs + bv;
            }
        }
    }
}

template <int W>
static void launch_group(const float* x, const BandArgs& args, float* inv,
                         float* invmu, float* rowsum, float* out, int bandBase,
                         int count, hipStream_t stream) {
    dim3 gs(B_, count);
    band_stats<W><<<gs, 256, 0, stream>>>(x, inv, invmu, bandBase);
    band_rowsum<W><<<count, 128, 0, stream>>>(args, rowsum, bandBase);
    dim3 gg((MROWS + 16 * MTILE - 1) / (16 * MTILE), count);
    band_gemm<W><<<gg, 256, 0, stream>>>(x, args, inv, invmu, rowsum, out,
                                         bandBase);
}

extern "C" void kernel_launch(void* const* d_in, const int* in_sizes, int n_in,
                              void* d_out, int out_size, void* d_ws, size_t ws_size,
                              hipStream_t stream) {
    const float* x = (const float*)d_in[0];
    BandArgs args;
    for (int i = 0; i < NBANDS; ++i) {
        args.W[i] = (const float*)d_in[1 + i];
        args.bias[i] = (const float*)d_in[1 + NBANDS + i];
    }
    float* ws = (float*)d_ws;
    float* inv = ws;                       // 43*8
    float* invmu = ws + NBANDS * B_;       // 43*8
    float* rowsum = ws + 2 * NBANDS * B_;  // 43*128
    float* out = (float*)d_out;

    launch_group<10>(x, args, inv, invmu, rowsum, out, 0, 20, stream);
    launch_group<20>(x, args, inv, invmu, rowsum, out, 20, 10, stream);
    launch_group<40>(x, args, inv, invmu, rowsum, out, 30, 8, stream);
    launch_group<61>(x, args, inv, invmu, rowsum, out, 38, 5, stream);
}